// Select_Profile_53274774340166
// MI455X (gfx1250) — compile-verified
//
#include <hip/hip_runtime.h>
#include <stdint.h>

typedef uint32_t v4u __attribute__((ext_vector_type(4)));
typedef int      v8i __attribute__((ext_vector_type(8)));
typedef int      v4i __attribute__((ext_vector_type(4)));

#define VOL       256
#define TILE      8
#define NTHREADS  256

__global__ __launch_bounds__(NTHREADS) void psf_sample_kernel(
    const float* __restrict__ vol,        // (256,256,256)
    const float* __restrict__ sampleGrid, // (B,3)
    const float* __restrict__ ax,         // (B,1,6)
    const float* __restrict__ bound,      // (B,2,3)
    const float* __restrict__ invcov,     // (B,3,3)
    const float* __restrict__ xyz_psf,    // (B,n,3)
    float* __restrict__ out,              // (B,)
    int n)
{
    __shared__ float tile[TILE * TILE * TILE];
    __shared__ float redW[NTHREADS / 32];
    __shared__ float redV[NTHREADS / 32];

    const int b   = blockIdx.x;
    const int tid = threadIdx.x;

    // ---------- per-b uniform parameters (scalarized by the compiler) ----------
    const float vx = ax[b * 6 + 0], vy = ax[b * 6 + 1], vz = ax[b * 6 + 2];
    const float tx = ax[b * 6 + 3], ty = ax[b * 6 + 4], tz = ax[b * 6 + 5];

    const float theta = sqrtf(vx * vx + vy * vy + vz * vz + 1e-12f);
    const float it    = 1.0f / theta;
    const float kx = vx * it, ky = vy * it, kz = vz * it;
    const float sn = sinf(theta);
    const float cs = cosf(theta);
    const float oc = 1.0f - cs;

    // Rodrigues: R = cI + s K + (1-c) k k^T
    const float R00 = cs + oc * kx * kx, R01 = oc * kx * ky - sn * kz, R02 = oc * kx * kz + sn * ky;
    const float R10 = oc * kx * ky + sn * kz, R11 = cs + oc * ky * ky, R12 = oc * ky * kz - sn * kx;
    const float R20 = oc * kx * kz - sn * ky, R21 = oc * ky * kz + sn * kx, R22 = cs + oc * kz * kz;

    const float gx0 = sampleGrid[b * 3 + 0] + tx;
    const float gy0 = sampleGrid[b * 3 + 1] + ty;
    const float gz0 = sampleGrid[b * 3 + 2] + tz;

    const float ctx = R00 * gx0 + R01 * gy0 + R02 * gz0;
    const float cty = R10 * gx0 + R11 * gy0 + R12 * gz0;
    const float ctz = R20 * gx0 + R21 * gy0 + R22 * gz0;

    // grid_sample coordinate transform collapses to: i = p * VOL/(VOL-1) - 0.5
    const float SC  = (float)VOL / (float)(VOL - 1);
    const float icx = ctx * SC - 0.5f;
    const float icy = cty * SC - 0.5f;
    const float icz = ctz * SC - 0.5f;

    // half-extent (|bound_psf| < 2.5 voxels -> 8-wide tile suffices)
    const float hx = (bound[b * 6 + 3] - bound[b * 6 + 0]) * 0.5f;
    const float hy = (bound[b * 6 + 4] - bound[b * 6 + 1]) * 0.5f;
    const float hz = (bound[b * 6 + 5] - bound[b * 6 + 2]) * 0.5f;

    const float M00 = invcov[b * 9 + 0], M01 = invcov[b * 9 + 1], M02 = invcov[b * 9 + 2];
    const float M10 = invcov[b * 9 + 3], M11 = invcov[b * 9 + 4], M12 = invcov[b * 9 + 5];
    const float M20 = invcov[b * 9 + 6], M21 = invcov[b * 9 + 7], M22 = invcov[b * 9 + 8];

    // tile base, clamped so the TDM tile never leaves the tensor
    int bx = (int)floorf(icx) - 3; bx = bx < 0 ? 0 : (bx > VOL - TILE ? VOL - TILE : bx);
    int by = (int)floorf(icy) - 3; by = by < 0 ? 0 : (by > VOL - TILE ? VOL - TILE : by);
    int bz = (int)floorf(icz) - 3; bz = bz < 0 ? 0 : (bz > VOL - TILE ? VOL - TILE : bz);
    bx = __builtin_amdgcn_readfirstlane(bx);
    by = __builtin_amdgcn_readfirstlane(by);
    bz = __builtin_amdgcn_readfirstlane(bz);

    // ---------- TDM: DMA the 8x8x8 float tile into LDS (wave 0 only) ----------
    if (tid < 32) {
        const uint32_t lds_off = (uint32_t)(uintptr_t)&tile[0]; // flat LDS addr low 32b == LDS offset
        const uint64_t ga = (uint64_t)(uintptr_t)(vol + ((size_t)bz * VOL + (size_t)by) * VOL + (size_t)bx);

        // D# group0: count=1 | lds_addr | global_addr[56:0] | type=2
        v4u g0 = { 1u,
                   lds_off,
                   (uint32_t)ga,
                   (uint32_t)((ga >> 32) & 0x1FFFFFFu) | (2u << 30) };
        // D# group1: wg_mask=0, data_size=2(4B); tensor_dim0/1=256; tile 8x8x8;
        //            dim0_stride=256, dim1_stride=65536
        v8i g1 = { (int)0x00020000u,          // data_size=4B at [17:16]
                   (int)(256u << 16),         // tensor_dim0[15:0] at [63:48]
                   (int)(256u << 16),         // tensor_dim1[15:0] at [95:80]
                   (int)(8u   << 16),         // tile_dim0 at [127:112]
                   (int)((8u << 16) | 8u),    // tile_dim1 | tile_dim2<<16
                   256,                       // tensor_dim0_stride[31:0]
                   0,                         // stride0[47:32]=0, stride1[15:0]=0
                   1 };                       // tensor_dim1_stride[47:16] (65536>>16)
        // D# group2: tensor_dim2=256, no iterate, dim2_stride=256^3 (unused for 3D tile)
        v4i g2 = { 256, 0, 0x01000000, 0 };
        v4i g3 = { 0, 0, 0, 0 };
        v8i g4 = { 0, 0, 0, 0, 0, 0, 0, 0 }; // extra group on clang-23 lane (zero-filled)

        __builtin_amdgcn_tensor_load_to_lds(g0, g1, g2, g3, g4, 0);
        __builtin_amdgcn_s_wait_tensorcnt(0);
    }
    __syncthreads();

    // ---------- per-sample loop: all gathers hit the LDS tile ----------
    const float* __restrict__ psf = xyz_psf + (size_t)b * (size_t)n * 3;
    float sw = 0.0f, svw = 0.0f;

    for (int i = tid; i < n; i += NTHREADS) {
        const float u0 = __builtin_nontemporal_load(&psf[i * 3 + 0]);
        const float u1 = __builtin_nontemporal_load(&psf[i * 3 + 1]);
        const float u2 = __builtin_nontemporal_load(&psf[i * 3 + 2]);

        // 2*sigmoid(u)-1
        const float s0 = 2.0f / (1.0f + expf(-u0)) - 1.0f;
        const float s1 = 2.0f / (1.0f + expf(-u1)) - 1.0f;
        const float s2 = 2.0f / (1.0f + expf(-u2)) - 1.0f;

        const float bpx = s0 * hx, bpy = s1 * hy, bpz = s2 * hz;

        // Gaussian weight: q = bp^T M bp
        const float q = bpx * (M00 * bpx + M01 * bpy + M02 * bpz)
                      + bpy * (M10 * bpx + M11 * bpy + M12 * bpz)
                      + bpz * (M20 * bpx + M21 * bpy + M22 * bpz);
        const float w = expf(-0.5f * q);

        const float ix = icx + bpx * SC;
        const float iy = icy + bpy * SC;
        const float iz = icz + bpz * SC;

        const float fxf = floorf(ix), fyf = floorf(iy), fzf = floorf(iz);
        const float fx = ix - fxf, fy = iy - fyf, fz = iz - fzf;
        const int x0 = (int)fxf, y0 = (int)fyf, z0 = (int)fzf;

        float v = 0.0f;
        #pragma unroll
        for (int dz = 0; dz < 2; ++dz) {
            const float wz = dz ? fz : 1.0f - fz;
            const int gzc = z0 + dz;
            #pragma unroll
            for (int dy = 0; dy < 2; ++dy) {
                const float wy = dy ? fy : 1.0f - fy;
                const int gyc = y0 + dy;
                #pragma unroll
                for (int dx = 0; dx < 2; ++dx) {
                    const float wx = dx ? fx : 1.0f - fx;
                    const int gxc = x0 + dx;
                    const bool ok = ((unsigned)gxc < (unsigned)VOL) &
                                    ((unsigned)gyc < (unsigned)VOL) &
                                    ((unsigned)gzc < (unsigned)VOL);
                    // masked local index keeps LDS read in-bounds; invalid corners add 0
                    const int li = (((gzc - bz) & 7) * TILE + ((gyc - by) & 7)) * TILE
                                 + ((gxc - bx) & 7);
                    const float val = tile[li];
                    v += ok ? (wx * wy * wz) * val : 0.0f;
                }
            }
        }
        sw  += w;
        svw += v * w;
    }

    // ---------- reduction: wave32 shuffle, then 8 waves via LDS ----------
    #pragma unroll
    for (int off = 16; off > 0; off >>= 1) {
        sw  += __shfl_xor(sw,  off, 32);
        svw += __shfl_xor(svw, off, 32);
    }
    const int wave = tid >> 5;
    if ((tid & 31) == 0) { redW[wave] = sw; redV[wave] = svw; }
    __syncthreads();
    if (tid == 0) {
        float SW = 0.0f, SV = 0.0f;
        #pragma unroll
        for (int k = 0; k < NTHREADS / 32; ++k) { SW += redW[k]; SV += redV[k]; }
        out[b] = SV / SW;
    }
}

extern "C" void kernel_launch(void* const* d_in, const int* in_sizes, int n_in,
                              void* d_out, int out_size, void* d_ws, size_t ws_size,
                              hipStream_t stream) {
    const float* x          = (const float*)d_in[0]; // (1,1,256,256,256)
    const float* sampleGrid = (const float*)d_in[1]; // (B,3)
    const float* ax         = (const float*)d_in[2]; // (B,1,6)
    const float* bound      = (const float*)d_in[3]; // (B,2,3)
    const float* invcov     = (const float*)d_in[4]; // (B,3,3)
    // d_in[5] = psf_sigma (unused by the reference)
    const float* xyz_psf    = (const float*)d_in[6]; // (B,n,3)
    float* out = (float*)d_out;

    const int B = in_sizes[1] / 3;
    const int n = (int)((long long)in_sizes[6] / (3LL * B));

    psf_sample_kernel<<<B, NTHREADS, 0, stream>>>(x, sampleGrid, ax, bound, invcov,
                                                  xyz_psf, out, n);
}